// GaussianKDE_39273180954692
// MI455X (gfx1250) — compile-verified
//
#include <hip/hip_runtime.h>

typedef __attribute__((ext_vector_type(2))) float v2f;
typedef __attribute__((ext_vector_type(8))) float v8f;

#define QDIM 4096
#define NDIM 50000
#define DDIM 64
#define MTILE 64
#define NTILE 32
#define NSPLIT 16
#define SF_STRIDE (DDIM + 2)   // pad LDS rows: stride 264B avoids 64-bank conflicts

// tiles per split: ceil(ceil(50000/32)/16) = ceil(1563/16) = 98 (even -> clean 2x pipeline)
#define TPS 98

// ---------------------------------------------------------------------------
// Prep 1: f = features @ bw_transform (one row per block, 64 threads), plus f2.
// ---------------------------------------------------------------------------
__global__ void kde_prep_f(const float* __restrict__ features,
                           const float* __restrict__ bw,   // [D,D] row-major
                           float* __restrict__ f,          // [Q,D]
                           float* __restrict__ f2)         // [Q]
{
    __shared__ float srow[DDIM];
    __shared__ float sred[DDIM];
    const int q = blockIdx.x;
    const int t = threadIdx.x;          // 0..63
    srow[t] = features[q * DDIM + t];
    __syncthreads();
    float acc = 0.0f;
    #pragma unroll 8
    for (int k = 0; k < DDIM; ++k)
        acc = fmaf(srow[k], bw[k * DDIM + t], acc);
    f[q * DDIM + t] = acc;
    sred[t] = acc * acc;
    __syncthreads();
    for (int s = 32; s > 0; s >>= 1) {
        if (t < s) sred[t] += sred[t + s];
        __syncthreads();
    }
    if (t == 0) f2[q] = sred[0];
}

// ---------------------------------------------------------------------------
// Prep 2: x2[n] = ||dataset[n]||^2. One wave32 per row (2 floats/lane).
// ---------------------------------------------------------------------------
__global__ void kde_x2(const float* __restrict__ dataset, float* __restrict__ x2)
{
    const int w    = (blockIdx.x * blockDim.x + threadIdx.x) >> 5;
    const int lane = threadIdx.x & 31;
    if (w >= NDIM) return;
    v2f d = *(const v2f*)(dataset + w * DDIM + lane * 2);
    float s = d.x * d.x + d.y * d.y;
    s += __shfl_xor(s, 1, 32);
    s += __shfl_xor(s, 2, 32);
    s += __shfl_xor(s, 4, 32);
    s += __shfl_xor(s, 8, 32);
    s += __shfl_xor(s, 16, 32);
    if (lane == 0) x2[w] = s;
}

__global__ void kde_zero(float* __restrict__ out)
{
    const int i = blockIdx.x * blockDim.x + threadIdx.x;
    if (i < QDIM) out[i] = 0.0f;
}

// ---------------------------------------------------------------------------
// Main fused kernel: WMMA fp32 GEMM tile -> exp -> reduce over N.
// 256 threads = 8 waves; wave (wm,wn) owns a 16x16 C tile of the 64x32 block
// tile. Double-buffered B fragments pipeline global loads against the WMMA
// chain. Grid: (Q/64, NSPLIT); partials merged with ds/global atomics.
// ---------------------------------------------------------------------------
__global__ __launch_bounds__(256)
void kde_main(const float* __restrict__ f,        // [Q,D]
              const float* __restrict__ f2,       // [Q]
              const float* __restrict__ dataset,  // [N,D]
              const float* __restrict__ x2,       // [N]
              const float* __restrict__ norm,     // [1]
              float* __restrict__ out)            // [Q]
{
    __shared__ float sf[MTILE * SF_STRIDE];
    __shared__ float srow[MTILE];

    const int qbase = blockIdx.x * MTILE;
    const int tid   = threadIdx.x;
    const int wave  = tid >> 5;
    const int lane  = tid & 31;

    // Cooperative load of the A tile (f rows) into padded LDS.
    {
        const v2f* src = (const v2f*)(f + qbase * DDIM);
        for (int i = tid; i < MTILE * (DDIM / 2); i += 256) {
            int row = i >> 5;
            int c   = i & 31;
            *(v2f*)(sf + row * SF_STRIDE + c * 2) = src[row * 32 + c];
        }
    }
    if (tid < MTILE) srow[tid] = 0.0f;
    __syncthreads();

    const int wm    = wave >> 1;      // 0..3 : M sub-tile
    const int wn    = wave & 1;       // 0..1 : N sub-tile
    const int lhalf = lane >> 4;      // lane half -> K pair select
    const int l15   = lane & 15;

    // A fragments: VGPR j of step k holds K = 4k + j + 2*lhalf, M = l15.
    v2f aFrag[16];
    {
        const float* aRowBase = sf + (wm * 16 + l15) * SF_STRIDE + lhalf * 2;
        #pragma unroll
        for (int k = 0; k < 16; ++k)
            aFrag[k] = *(const v2f*)(aRowBase + 4 * k);
    }

    const int t0 = blockIdx.y * TPS;
    float lsum[8] = {0.f, 0.f, 0.f, 0.f, 0.f, 0.f, 0.f, 0.f};

    // B fragment loader: clamp address, mask handles both N-edge and overrun
    // tiles (any tile index >= 1563 has all cols >= N).
    #define LOADB(TT, B, X2V, MV)                                            \
        do {                                                                 \
            int col_  = (TT) * NTILE + wn * 16 + l15;                        \
            int colc_ = (col_ < NDIM) ? col_ : (NDIM - 1);                   \
            const float* bb_ = dataset + colc_ * DDIM + lhalf * 2;           \
            _Pragma("unroll")                                                \
            for (int k_ = 0; k_ < 16; ++k_)                                  \
                (B)[k_] = *(const v2f*)(bb_ + 4 * k_);                       \
            (X2V) = x2[colc_];                                               \
            (MV)  = (col_ < NDIM) ? 1.0f : 0.0f;                             \
        } while (0)

    #define COMPUTE(B, X2V, MV)                                              \
        do {                                                                 \
            v8f acc_ = {0.f, 0.f, 0.f, 0.f, 0.f, 0.f, 0.f, 0.f};             \
            _Pragma("unroll")                                                \
            for (int k_ = 0; k_ < 16; ++k_)                                  \
                acc_ = __builtin_amdgcn_wmma_f32_16x16x4_f32(                \
                           false, aFrag[k_], false, (B)[k_],                 \
                           (short)0, acc_, false, false);                    \
            _Pragma("unroll")                                                \
            for (int v_ = 0; v_ < 8; ++v_)                                   \
                lsum[v_] += __expf(acc_[v_] - 0.5f * (X2V)) * (MV);          \
        } while (0)

    v2f b0[16], b1[16];
    float x2v0, m0, x2v1, m1;

    LOADB(t0, b0, x2v0, m0);
    for (int i = 0; i < TPS; i += 2) {
        LOADB(t0 + i + 1, b1, x2v1, m1);   // prefetch odd tile
        COMPUTE(b0, x2v0, m0);             // consume even tile
        LOADB(t0 + i + 2, b0, x2v0, m0);   // prefetch next even tile
        COMPUTE(b1, x2v1, m1);             // consume odd tile
    }

    #undef LOADB
    #undef COMPUTE

    // Reduce lsum over the 16 N-columns held in each lane half.
    #pragma unroll
    for (int v = 0; v < 8; ++v) {
        float s = lsum[v];
        s += __shfl_xor(s, 1, 32);
        s += __shfl_xor(s, 2, 32);
        s += __shfl_xor(s, 4, 32);
        s += __shfl_xor(s, 8, 32);
        lsum[v] = s;
    }
    if (l15 == 0) {
        #pragma unroll
        for (int v = 0; v < 8; ++v)
            atomicAdd(&srow[wm * 16 + v + lhalf * 8], lsum[v]);  // ds_add_f32
    }
    __syncthreads();

    if (tid < MTILE) {
        const int q = qbase + tid;
        const float scale = norm[0] * (1.0f / (float)NDIM) * __expf(-0.5f * f2[q]);
        atomicAdd(out + q, srow[tid] * scale);
    }
}

// ---------------------------------------------------------------------------
// Launch. Workspace layout: f (Q*D) | f2 (Q) | x2 (N) -> ~1.27 MB.
// ---------------------------------------------------------------------------
extern "C" void kernel_launch(void* const* d_in, const int* in_sizes, int n_in,
                              void* d_out, int out_size, void* d_ws, size_t ws_size,
                              hipStream_t stream)
{
    const float* features = (const float*)d_in[0];
    const float* bw       = (const float*)d_in[1];
    const float* dataset  = (const float*)d_in[2];
    const float* norm     = (const float*)d_in[3];
    float* out = (float*)d_out;

    float* f  = (float*)d_ws;
    float* f2 = f + QDIM * DDIM;
    float* x2 = f2 + QDIM;

    kde_prep_f<<<QDIM, 64, 0, stream>>>(features, bw, f, f2);
    kde_x2<<<(NDIM + 7) / 8, 256, 0, stream>>>(dataset, x2);
    kde_zero<<<(QDIM + 255) / 256, 256, 0, stream>>>(out);

    dim3 grid(QDIM / MTILE, NSPLIT);
    kde_main<<<grid, 256, 0, stream>>>(f, f2, dataset, x2, norm, out);
}